// GraphAttentionNetwork_738734375740
// MI455X (gfx1250) — compile-verified
//
#include <hip/hip_runtime.h>
#include <math.h>

// ---------------------------------------------------------------------------
// GAT forward for MI455X (gfx1250, wave32).
// - All GEMMs (node_proj/res_proj/out_proj/message-aggregation) and the
//   per-edge MLP use V_WMMA_F32_16X16X4_F32 (fp32 in, fp32 acc).
// - Edge tensor (100MB) processed by one fused WMMA kernel per layer:
//   LN -> edge1(12->24) -> gelu -> edge2(24->12) -> +residual -> le dot.
// - gelu in the edge kernel uses a branchless A&S erf approximation
//   (|err| <= 1.5e-7) with hardware v_rcp_f32 / v_exp_f32: libm erff lowers
//   to a divergent two-path polynomial, and __frcp_rn lowers to a full IEEE
//   division sequence -- both too costly at 50M calls/layer.
// ---------------------------------------------------------------------------

#define BATCH 8
#define NNODE 512
#define EF    12
#define FOUT  128
#define MROWS (BATCH * NNODE) // 4096

typedef float v2f __attribute__((ext_vector_type(2)));
typedef float v8f __attribute__((ext_vector_type(8)));

__device__ __forceinline__ float gelu_exact(float x) {
    return 0.5f * x * (1.0f + erff(x * 0.70710678118654752f));
}
// Branchless erf (Abramowitz-Stegun 7.1.26, |abs err| <= 1.5e-7).
// __builtin_amdgcn_rcpf -> single v_rcp_f32 (fast HW reciprocal).
__device__ __forceinline__ float erf_fast(float y) {
    float ay = fabsf(y);
    float t = __builtin_amdgcn_rcpf(1.f + 0.3275911f * ay);
    float poly = t * (0.254829592f +
                 t * (-0.284496736f +
                 t * (1.421413741f +
                 t * (-1.453152027f +
                 t * 1.061405429f))));
    float r = 1.f - poly * __expf(-ay * ay);
    return copysignf(r, y);
}
__device__ __forceinline__ float gelu_fast(float x) {
    return 0.5f * x * (1.0f + erf_fast(x * 0.70710678118654752f));
}
__device__ __forceinline__ float leaky02(float x) { return x > 0.f ? x : 0.2f * x; }

__device__ __forceinline__ v8f wmma_f32_16x16x4(v2f a, v2f b, v8f c) {
    // (neg_a, A, neg_b, B, c_mod, C, reuse_a, reuse_b)
    return __builtin_amdgcn_wmma_f32_16x16x4_f32(false, a, false, b, (short)0, c,
                                                 false, false);
}

// ---------------------------------------------------------------------------
// a_sum[j] = sum_h attn[h][j]   (dim = 2*hs + EF)
// ---------------------------------------------------------------------------
__global__ void k_asum(const float* __restrict__ attn, int heads, int dim,
                       float* __restrict__ asum) {
    int j = blockIdx.x * blockDim.x + threadIdx.x;
    if (j >= dim) return;
    float s = 0.f;
    for (int h = 0; h < heads; ++h) s += attn[h * dim + j];
    asum[j] = s;
}

// ---------------------------------------------------------------------------
// Fused WMMA edge pipeline. Block = 256 threads = 8 waves; each wave owns 32
// edges (block: 256 edges). In-place (src==dst) is safe: each block reads only
// its own 256 edges before writing them.
// ---------------------------------------------------------------------------
__global__ void k_edge_wmma(const float* __restrict__ src, float* __restrict__ dst,
                            float* __restrict__ le,
                            const float* __restrict__ W1, const float* __restrict__ b1,
                            const float* __restrict__ W2, const float* __restrict__ b2,
                            const float* __restrict__ g,  const float* __restrict__ bl,
                            const float* __restrict__ aedge) {
    __shared__ float sW1[24 * 12], sW2[12 * 24];
    __shared__ float sB1[24], sB2[12], sG[12], sBl[12], sAe[12];
    __shared__ __align__(16) float sE[8][32][EF];   // original e -> new_edge (in place)
    __shared__ __align__(16) float sLN[8][32][EF];  // eln
    __shared__ __align__(16) float sH1[8][32][24];  // gelu(edge1)

    int tid = threadIdx.x;
    int wave = tid >> 5, lane = tid & 31;
    size_t ebase = (size_t)blockIdx.x * 256 + (size_t)wave * 32;

    for (int i = tid; i < 288; i += 256) { sW1[i] = W1[i]; sW2[i] = W2[i]; }
    if (tid < 24) sB1[tid] = b1[tid];
    if (tid < 12) {
        sB2[tid] = b2[tid];
        sG[tid]  = g[tid];
        sBl[tid] = bl[tid];
        sAe[tid] = aedge[tid];
    }

    // ---- stage 1: load + LayerNorm (one edge per lane) ----
    size_t eidx = ebase + lane;
    const float4* ep4 = (const float4*)(src + eidx * EF); // 48B rows, 16B aligned
    float4 q0 = ep4[0], q1 = ep4[1], q2 = ep4[2];
    float e[EF] = {q0.x, q0.y, q0.z, q0.w, q1.x, q1.y, q1.z, q1.w,
                   q2.x, q2.y, q2.z, q2.w};
    float mu = 0.f;
#pragma unroll
    for (int k = 0; k < EF; ++k) mu += e[k];
    mu *= (1.f / EF);
    float var = 0.f;
#pragma unroll
    for (int k = 0; k < EF; ++k) { float d = e[k] - mu; var += d * d; }
    var *= (1.f / EF);
    float rinv = rsqrtf(var + 1e-5f);
    __syncthreads(); // weights ready
#pragma unroll
    for (int k = 0; k < EF; ++k) {
        sE[wave][lane][k]  = e[k];
        sLN[wave][lane][k] = (e[k] - mu) * rinv * sG[k] + sBl[k];
    }
    __syncthreads();

    int l15 = lane & 15;
    int kHi = (lane >> 4) * 2;
    int rbase = (lane >> 4) * 8; // C/D rows: v + 8*(lane>=16)

    // ---- stage 2: edge1 (12 -> 24) + gelu ----
#pragma unroll
    for (int t = 0; t < 2; ++t) {
#pragma unroll
        for (int j0 = 0; j0 < 2; ++j0) {
            int col = j0 * 16 + l15; // valid < 24
            v8f acc = {};
#pragma unroll
            for (int k0 = 0; k0 < 12; k0 += 4) {
                v2f a, b;
                a.x = sLN[wave][t * 16 + l15][k0 + kHi];
                a.y = sLN[wave][t * 16 + l15][k0 + kHi + 1];
                b.x = (col < 24) ? sW1[col * 12 + k0 + kHi] : 0.f;
                b.y = (col < 24) ? sW1[col * 12 + k0 + kHi + 1] : 0.f;
                acc = wmma_f32_16x16x4(a, b, acc);
            }
            if (col < 24) {
                float bb = sB1[col];
#pragma unroll
                for (int v = 0; v < 8; ++v)
                    sH1[wave][t * 16 + rbase + v][col] = gelu_fast(acc[v] + bb);
            }
        }
    }
    __syncthreads();

    // ---- stage 3: edge2 (24 -> 12) + bias + residual (in place into sE) ----
#pragma unroll
    for (int t = 0; t < 2; ++t) {
        int col = l15; // valid < 12
        v8f acc = {};
#pragma unroll
        for (int k0 = 0; k0 < 24; k0 += 4) {
            v2f a, b;
            a.x = sH1[wave][t * 16 + l15][k0 + kHi];
            a.y = sH1[wave][t * 16 + l15][k0 + kHi + 1];
            b.x = (col < 12) ? sW2[col * 24 + k0 + kHi] : 0.f;
            b.y = (col < 12) ? sW2[col * 24 + k0 + kHi + 1] : 0.f;
            acc = wmma_f32_16x16x4(a, b, acc);
        }
        if (col < 12) {
            float bb = sB2[col];
#pragma unroll
            for (int v = 0; v < 8; ++v) {
                int r = t * 16 + rbase + v;
                sE[wave][r][col] = acc[v] + bb + sE[wave][r][col];
            }
        }
    }
    __syncthreads();

    // ---- stage 4: coalesced store of new_edge + le ----
    float4* op4 = (float4*)(dst + ebase * EF);
    const float4* so4 = (const float4*)&sE[wave][0][0]; // 32*12 = 96 float4
#pragma unroll
    for (int i = 0; i < 3; ++i) op4[lane + i * 32] = so4[lane + i * 32];
    float lev = 0.f;
#pragma unroll
    for (int k = 0; k < EF; ++k) lev += sE[wave][lane][k] * sAe[k];
    le[eidx] = lev;
}

// ---------------------------------------------------------------------------
// Row LayerNorm: y[row, t] = (x-mu)*rsqrt(var+eps)*g + b, one block per row.
// ---------------------------------------------------------------------------
__global__ void k_ln_rows(const float* __restrict__ x, const float* __restrict__ g,
                          const float* __restrict__ b, float* __restrict__ y, int K) {
    __shared__ float red[128];
    int row = blockIdx.x, t = threadIdx.x;
    float v = (t < K) ? x[(size_t)row * K + t] : 0.f;
    red[t] = v;
    __syncthreads();
    for (int s = 64; s > 0; s >>= 1) { if (t < s) red[t] += red[t + s]; __syncthreads(); }
    float mu = red[0] / K;
    __syncthreads();
    float d = (t < K) ? v - mu : 0.f;
    red[t] = d * d;
    __syncthreads();
    for (int s = 64; s > 0; s >>= 1) { if (t < s) red[t] += red[t + s]; __syncthreads(); }
    float var = red[0] / K;
    if (t < K) y[(size_t)row * K + t] = d * rsqrtf(var + 1e-5f) * g[t] + b[t];
}

// ---------------------------------------------------------------------------
// Y[M,Nout] = act(X[M,K] @ W[Nout,K]^T + bias (+R)), fp32 WMMA 16x16x4.
// 8 waves/block, one 16x16 tile per wave; grid.x = M/16 (Nout==128).
// ---------------------------------------------------------------------------
__global__ void k_gemm_wmma(const float* __restrict__ X, const float* __restrict__ W,
                            const float* __restrict__ bias, const float* __restrict__ R,
                            float* __restrict__ Y, int K, int Nout, int leaky) {
    int lane = threadIdx.x & 31;
    int wave = threadIdx.x >> 5;
    int row0 = blockIdx.x * 16;
    int col0 = wave * 16;
    if (col0 >= Nout) return; // wave-uniform
    int l15 = lane & 15;
    int kHi = (lane >> 4) * 2;
    const float* xr = X + (size_t)(row0 + l15) * K;
    const float* wr = W + (size_t)(col0 + l15) * K;
    v8f acc = {};
    for (int k0 = 0; k0 < K; k0 += 4) {
        v2f a, b;
        a.x = xr[k0 + kHi]; a.y = xr[k0 + kHi + 1];
        b.x = wr[k0 + kHi]; b.y = wr[k0 + kHi + 1];
        acc = wmma_f32_16x16x4(a, b, acc);
    }
    int col = col0 + l15;
    float bv = bias[col];
    int rbase = row0 + (lane >> 4) * 8;
#pragma unroll
    for (int v = 0; v < 8; ++v) {
        size_t oi = (size_t)(rbase + v) * Nout + col;
        float val = acc[v] + bv;
        if (R) val += R[oi];
        if (leaky) val = leaky02(val);
        Y[oi] = val;
    }
}

// ---------------------------------------------------------------------------
// lr[bn,h] = dot(h[bn, h*hs:], a_row);  lc = same with a_col (a_sum[hs:2hs]).
// ---------------------------------------------------------------------------
__global__ void k_lrlc(const float* __restrict__ h, const float* __restrict__ asum,
                       float* __restrict__ lr, float* __restrict__ lc,
                       int heads, int hs) {
    int idx = blockIdx.x * blockDim.x + threadIdx.x;
    if (idx >= MROWS * heads) return;
    int hIdx = idx % heads;
    int bn = idx / heads;
    const float* hp = h + (size_t)bn * FOUT + hIdx * hs;
    float r = 0.f, c = 0.f;
    for (int d = 0; d < hs; ++d) { float v = hp[d]; r += v * asum[d]; c += v * asum[hs + d]; }
    lr[idx] = r;
    lc[idx] = c;
}

// ---------------------------------------------------------------------------
// Attention softmax over m (len 512): one wave32 per (b,n,h); 16 m per lane in
// registers; shuffle reductions. coeffs written as (b, h, n, m) so the msg
// GEMM A-operand is contiguous. All-masked rows -> zeros (nan_to_num).
// ---------------------------------------------------------------------------
__global__ void k_attn_softmax(const float* __restrict__ lr, const float* __restrict__ lc,
                               const float* __restrict__ le, const int* __restrict__ adj,
                               float* __restrict__ coeffs, int heads) {
    int wave = threadIdx.x >> 5, lane = threadIdx.x & 31;
    int gw = blockIdx.x * (blockDim.x >> 5) + wave; // (b*N+n)*heads + h
    int hIdx = gw % heads;
    int bn = gw / heads;
    int b = bn / NNODE;
    int n = bn - b * NNODE;
    float lrv = lr[gw];
    const int*   ar  = adj + (size_t)bn * NNODE;
    const float* ler = le + (size_t)bn * NNODE;
    const float* lcb = lc + (size_t)b * NNODE * heads + hIdx;

    float vals[16];
    float mx = -INFINITY;
#pragma unroll
    for (int i = 0; i < 16; ++i) {
        int m = lane + i * 32;
        float lg = -INFINITY;
        if (ar[m] != 0) {
            lg = leaky02(lrv + lcb[(size_t)m * heads] + ler[m]);
        }
        vals[i] = lg;
        mx = fmaxf(mx, lg);
    }
    for (int off = 16; off > 0; off >>= 1) mx = fmaxf(mx, __shfl_xor(mx, off, 32));
    float s = 0.f;
#pragma unroll
    for (int i = 0; i < 16; ++i) {
        float ev = (vals[i] == -INFINITY) ? 0.f : __expf(vals[i] - mx);
        vals[i] = ev;
        s += ev;
    }
    for (int off = 16; off > 0; off >>= 1) s += __shfl_xor(s, off, 32);
    float invs = (s > 0.f) ? 1.f / s : 0.f;
    float* cr = coeffs + ((size_t)(b * heads + hIdx) * NNODE + n) * NNODE;
#pragma unroll
    for (int i = 0; i < 16; ++i) cr[lane + i * 32] = vals[i] * invs;
}

// ---------------------------------------------------------------------------
// msg[b,n,h*hs+d] = sum_m coeffs[b,h,n,m] * h[b,m,h*hs+d]  (fp32 WMMA, K=m).
// One wave per 16x16 (n,d) tile; grid = (N/16, FOUT/16, B).
// ---------------------------------------------------------------------------
__global__ void k_msg_wmma(const float* __restrict__ coeffs, const float* __restrict__ h,
                           float* __restrict__ msg, int heads, int hs) {
    int lane = threadIdx.x & 31;
    int n0 = blockIdx.x * 16;
    int dtiles = hs >> 4;
    int hIdx = blockIdx.y / dtiles;
    int d0 = (blockIdx.y % dtiles) * 16;
    int b = blockIdx.z;
    int l15 = lane & 15;
    int kHi = (lane >> 4) * 2;
    const float* ar = coeffs + ((size_t)(b * heads + hIdx) * NNODE + n0 + l15) * NNODE;
    const float* br = h + (size_t)b * NNODE * FOUT + hIdx * hs + d0 + l15;
    v8f acc = {};
    for (int m0 = 0; m0 < NNODE; m0 += 4) {
        int m = m0 + kHi;
        v2f a, bb;
        a.x = ar[m];
        a.y = ar[m + 1];
        bb.x = br[(size_t)m * FOUT];
        bb.y = br[(size_t)(m + 1) * FOUT];
        acc = wmma_f32_16x16x4(a, bb, acc);
    }
    int col = hIdx * hs + d0 + l15;
    int rbase = n0 + (lane >> 4) * 8;
#pragma unroll
    for (int v = 0; v < 8; ++v)
        msg[(size_t)(b * NNODE + rbase + v) * FOUT + col] = acc[v];
}

// ---------------------------------------------------------------------------
// Pooling head, one block per batch (exact erff here: only 1K calls).
// ---------------------------------------------------------------------------
__global__ void k_pool(const float* __restrict__ node,
                       const float* __restrict__ aW, const float* __restrict__ aB,
                       const float* __restrict__ p1W, const float* __restrict__ p1b,
                       const float* __restrict__ lng, const float* __restrict__ lnb,
                       const float* __restrict__ p2W, const float* __restrict__ p2b,
                       float* __restrict__ out) {
    __shared__ float sc[NNODE];
    __shared__ float red[256];
    __shared__ float pooled[FOUT];
    __shared__ float hdn[FOUT];
    int b = blockIdx.x, t = threadIdx.x;
    const float* nb = node + (size_t)b * NNODE * FOUT;

    for (int n = t; n < NNODE; n += 256) {
        float s = aB[0];
        const float* np = nb + (size_t)n * FOUT;
        for (int f = 0; f < FOUT; ++f) s += np[f] * aW[f];
        sc[n] = s;
    }
    __syncthreads();
    float m = -INFINITY;
    for (int n = t; n < NNODE; n += 256) m = fmaxf(m, sc[n]);
    red[t] = m;
    __syncthreads();
    for (int s2 = 128; s2 > 0; s2 >>= 1) { if (t < s2) red[t] = fmaxf(red[t], red[t + s2]); __syncthreads(); }
    float mx = red[0];
    __syncthreads();
    float ls = 0.f;
    for (int n = t; n < NNODE; n += 256) { float ev = expf(sc[n] - mx); sc[n] = ev; ls += ev; }
    red[t] = ls;
    __syncthreads();
    for (int s2 = 128; s2 > 0; s2 >>= 1) { if (t < s2) red[t] += red[t + s2]; __syncthreads(); }
    float invs = 1.f / red[0];
    __syncthreads();

    if (t < FOUT) {
        float p = 0.f;
        for (int n = 0; n < NNODE; ++n) p += sc[n] * nb[(size_t)n * FOUT + t];
        pooled[t] = p * invs;
    }
    __syncthreads();

    float z = 0.f;
    if (t < FOUT) {
        z = p1b[t];
        for (int f = 0; f < FOUT; ++f) z += p1W[t * FOUT + f] * pooled[f];
    }
    red[t] = (t < FOUT) ? z : 0.f;
    __syncthreads();
    for (int s2 = 128; s2 > 0; s2 >>= 1) { if (t < s2) red[t] += red[t + s2]; __syncthreads(); }
    float mu = red[0] / FOUT;
    __syncthreads();
    float d = (t < FOUT) ? z - mu : 0.f;
    red[t] = d * d;
    __syncthreads();
    for (int s2 = 128; s2 > 0; s2 >>= 1) { if (t < s2) red[t] += red[t + s2]; __syncthreads(); }
    float var = red[0] / FOUT;
    __syncthreads();
    if (t < FOUT) hdn[t] = gelu_exact(d * rsqrtf(var + 1e-5f) * lng[t] + lnb[t]);
    __syncthreads();

    red[t] = (t < FOUT) ? p2W[t] * hdn[t] : 0.f;
    __syncthreads();
    for (int s2 = 128; s2 > 0; s2 >>= 1) { if (t < s2) red[t] += red[t + s2]; __syncthreads(); }
    if (t == 0) {
        float y = red[0] + p2b[0];
        out[b] = 14.f / (1.f + expf(-y));
    }
}

// ---------------------------------------------------------------------------
// Host orchestration. d_in order (jax tree flatten, dicts sorted by key):
//  0 node, 1 edge, 2 adj,
//  layer bases {3,18,31}: attn, e1W,e1b, e2W,e2b, ln1_b,ln1_g, ln2_b,ln2_g,
//                         npW,npb, opW,opb, (rpW,rpb layer0 only)
//  pool: 44 attnW, 45 attnb, 46 ln_b, 47 ln_g, 48 p1W, 49 p1b, 50 p2W, 51 p2b
// ---------------------------------------------------------------------------
extern "C" void kernel_launch(void* const* d_in, const int* in_sizes, int n_in,
                              void* d_out, int out_size, void* d_ws, size_t ws_size,
                              hipStream_t stream) {
    (void)in_sizes; (void)n_in; (void)out_size; (void)ws_size;
    const float* node_in = (const float*)d_in[0];
    const float* edge_in = (const float*)d_in[1];
    const int*   adj     = (const int*)d_in[2];

    const size_t BNN = (size_t)BATCH * NNODE * NNODE;
    float* ws = (float*)d_ws;
    size_t o = 0;
    float* edge_ws = ws + o; o += BNN * EF;
    float* le      = ws + o; o += BNN;
    float* coeffs  = ws + o; o += BNN * 4;
    float* xln     = ws + o; o += (size_t)MROWS * FOUT;
    float* hbuf    = ws + o; o += (size_t)MROWS * FOUT;
    float* nodeA   = ws + o; o += (size_t)MROWS * FOUT;
    float* nodeB   = ws + o; o += (size_t)MROWS * FOUT;
    float* resb    = ws + o; o += (size_t)MROWS * FOUT;
    float* msg     = ws + o; o += (size_t)MROWS * FOUT;
    float* lrb     = ws + o; o += (size_t)MROWS * 4;
    float* lcb     = ws + o; o += (size_t)MROWS * 4;
    float* asum    = ws + o; o += 512;

    const int base[3]   = {3, 18, 31};
    const int headsA[3] = {4, 4, 1};
    const int finA[3]   = {64, 128, 128};
    const int leakyA[3] = {1, 1, 0};

    const float* cur = node_in;
    float* nextbuf = nodeA;
    const float* esrc = edge_in;

    for (int l = 0; l < 3; ++l) {
        int bi = base[l];
        const float* attn = (const float*)d_in[bi + 0];
        const float* e1W  = (const float*)d_in[bi + 1];
        const float* e1b  = (const float*)d_in[bi + 2];
        const float* e2W  = (const float*)d_in[bi + 3];
        const float* e2b  = (const float*)d_in[bi + 4];
        const float* ln1b = (const float*)d_in[bi + 5];
        const float* ln1g = (const float*)d_in[bi + 6];
        const float* ln2b = (const float*)d_in[bi + 7];
        const float* ln2g = (const float*)d_in[bi + 8];
        const float* npW  = (const float*)d_in[bi + 9];
        const float* npb  = (const float*)d_in[bi + 10];
        const float* opW  = (const float*)d_in[bi + 11];
        const float* opb  = (const float*)d_in[bi + 12];

        int heads = headsA[l], fin = finA[l];
        int hs = FOUT / heads;
        int adim = 2 * hs + EF;

        k_asum<<<1, 512, 0, stream>>>(attn, heads, adim, asum);
        k_edge_wmma<<<(unsigned)(BNN / 256), 256, 0, stream>>>(
            esrc, edge_ws, le, e1W, e1b, e2W, e2b, ln2g, ln2b, asum + 2 * hs);
        k_ln_rows<<<MROWS, 128, 0, stream>>>(cur, ln1g, ln1b, xln, fin);
        k_gemm_wmma<<<MROWS / 16, 256, 0, stream>>>(xln, npW, npb, nullptr, hbuf,
                                                    fin, FOUT, 0);
        if (l == 0) {
            const float* rpW = (const float*)d_in[bi + 13];
            const float* rpb = (const float*)d_in[bi + 14];
            k_gemm_wmma<<<MROWS / 16, 256, 0, stream>>>(cur, rpW, rpb, nullptr, resb,
                                                        fin, FOUT, 0);
        }
        k_lrlc<<<(MROWS * heads + 255) / 256, 256, 0, stream>>>(hbuf, asum, lrb, lcb,
                                                                heads, hs);
        k_attn_softmax<<<(MROWS * heads) / 8, 256, 0, stream>>>(lrb, lcb, le, adj,
                                                                coeffs, heads);
        k_msg_wmma<<<dim3(NNODE / 16, FOUT / 16, BATCH), 32, 0, stream>>>(
            coeffs, hbuf, msg, heads, hs);
        const float* resid = (l == 0) ? (const float*)resb : cur;
        k_gemm_wmma<<<MROWS / 16, 256, 0, stream>>>(msg, opW, opb, resid, nextbuf,
                                                    FOUT, FOUT, leakyA[l]);
        cur = nextbuf;
        nextbuf = (nextbuf == nodeA) ? nodeB : nodeA;
        esrc = edge_ws;
    }

    k_pool<<<BATCH, 256, 0, stream>>>(
        cur, (const float*)d_in[44], (const float*)d_in[45],
        (const float*)d_in[48], (const float*)d_in[49],
        (const float*)d_in[47], (const float*)d_in[46],
        (const float*)d_in[50], (const float*)d_in[51], (float*)d_out);
}